// ConvolutionFeatureModel_34325378630130
// MI455X (gfx1250) — compile-verified
//
#include <hip/hip_runtime.h>

typedef __attribute__((ext_vector_type(2))) float v2f;
typedef __attribute__((ext_vector_type(8))) float v8f;
typedef int v4i __attribute__((vector_size(4 * sizeof(int))));

#define BATCH 16384
#define IDIM  64
#define WIDTH 4096
#define BS_STRIDE 68   // padded LDS row stride (floats): conflict-free ds_load_b64

#if defined(__has_builtin)
#  if __has_builtin(__builtin_amdgcn_global_load_async_to_lds_b128)
#    define HAVE_ASYNC_LDS 1
#  endif
#  if __has_builtin(__builtin_amdgcn_s_wait_asynccnt)
#    define HAVE_WAIT_ASYNC 1
#  endif
#endif

// Address-space-typed pointers for the async global->LDS builtin.
typedef __attribute__((address_space(1))) const void* gas_cptr;
typedef __attribute__((address_space(3))) void*       las_ptr;
typedef __attribute__((address_space(1))) v4i*        gv4i_p;
typedef __attribute__((address_space(3))) v4i*        lv4i_p;

// ---------------------------------------------------------------------------
// Prologue: per-row squared L2 norms (x2[b], w2[w]) into workspace.
// ---------------------------------------------------------------------------
__global__ __launch_bounds__(256) void rownorm2_kernel(const float* __restrict__ src,
                                                       float* __restrict__ dst,
                                                       int nrows) {
    int r = blockIdx.x * blockDim.x + threadIdx.x;
    if (r >= nrows) return;
    const float4* p = (const float4*)(src + (size_t)r * IDIM);
    float s = 0.0f;
#pragma unroll
    for (int i = 0; i < IDIM / 4; ++i) {
        float4 v = p[i];
        s = fmaf(v.x, v.x, s);
        s = fmaf(v.y, v.y, s);
        s = fmaf(v.z, v.z, s);
        s = fmaf(v.w, v.w, s);
    }
    dst[r] = s;
}

// ---------------------------------------------------------------------------
// Main: xw GEMM via V_WMMA_F32_16X16X4_F32 + distance/GELU epilogue.
// Workgroup (8 waves): 128(M) x 64(N) of the output.
// B strip (64 weight rows) staged once per WG into LDS via async copy.
// ---------------------------------------------------------------------------
__global__ __launch_bounds__(256) void dist_gelu_wmma_kernel(
    const float* __restrict__ x,     // [BATCH, IDIM]
    const float* __restrict__ w,     // [WIDTH, IDIM]
    const float* __restrict__ x2,    // [BATCH]
    const float* __restrict__ w2,    // [WIDTH]
    float* __restrict__ out)         // [BATCH, WIDTH]
{
    __shared__ float bs[64 * BS_STRIDE];   // 17 KB (of 320 KB/WGP)

    const int tid  = threadIdx.x;
    const int lane = tid & 31;
    const int wave = tid >> 5;
    const int hi   = lane >> 4;      // 0: lanes 0-15, 1: lanes 16-31
    const int lo   = lane & 15;

    const int m0 = (blockIdx.y * 8 + wave) * 16;   // x rows for this wave
    const int n0 = blockIdx.x * 64;                // weight rows (out cols)

    // ---- Stage shared B strip: 64 rows x 64 floats -> LDS (padded rows) ----
    // 1024 16-byte chunks, 4 per thread.
#pragma unroll
    for (int i = 0; i < 4; ++i) {
        int c   = tid + 256 * i;           // 0..1023
        int row = c >> 4;                  // 0..63
        int col = (c & 15) * 4;            // 0..60
        const float* g = w + (size_t)(n0 + row) * IDIM + col;
        float*       l = &bs[row * BS_STRIDE + col];
#if defined(HAVE_ASYNC_LDS)
        __builtin_amdgcn_global_load_async_to_lds_b128(
            (gv4i_p)(gas_cptr)g, (lv4i_p)(las_ptr)l, 0, 0);
#else
        *(float4*)l = *(const float4*)g;
#endif
    }
#if defined(HAVE_ASYNC_LDS)
#  if defined(HAVE_WAIT_ASYNC)
    __builtin_amdgcn_s_wait_asynccnt(0);
#  else
    asm volatile("s_wait_asynccnt 0" ::: "memory");
#  endif
#endif
    __syncthreads();

    // ISA A(16x4) fp32 layout: lane holds (m=lo, k=4t+2*hi, +1) -> float2 loads.
    const float* xrow  = x + (size_t)(m0 + lo) * IDIM + 2 * hi;
    // B(4x16) fragment base in LDS for tile j: row (j*16+lo), k offset 2*hi.
    const int    bbase = lo * BS_STRIDE + 2 * hi;

    v8f acc[4] = {};

#pragma unroll
    for (int t = 0; t < IDIM / 4; ++t) {
        v2f a  = *(const v2f*)(xrow + 4 * t);
        v2f b0 = *(const v2f*)&bs[bbase + 0 * 16 * BS_STRIDE + 4 * t];
        v2f b1 = *(const v2f*)&bs[bbase + 1 * 16 * BS_STRIDE + 4 * t];
        v2f b2 = *(const v2f*)&bs[bbase + 2 * 16 * BS_STRIDE + 4 * t];
        v2f b3 = *(const v2f*)&bs[bbase + 3 * 16 * BS_STRIDE + 4 * t];
        acc[0] = __builtin_amdgcn_wmma_f32_16x16x4_f32(false, a, false, b0,
                                                       (short)0, acc[0], false, false);
        acc[1] = __builtin_amdgcn_wmma_f32_16x16x4_f32(false, a, false, b1,
                                                       (short)0, acc[1], false, false);
        acc[2] = __builtin_amdgcn_wmma_f32_16x16x4_f32(false, a, false, b2,
                                                       (short)0, acc[2], false, false);
        acc[3] = __builtin_amdgcn_wmma_f32_16x16x4_f32(false, a, false, b3,
                                                       (short)0, acc[3], false, false);
    }

    // C/D layout: vgpr r, lanes 0-15 -> (M=r, N=lo); lanes 16-31 -> (M=r+8, N=lo)
    const float4* px = (const float4*)(x2 + m0 + 8 * hi);   // 32B aligned
    float4 xa = px[0], xb = px[1];
    float x2r[8] = {xa.x, xa.y, xa.z, xa.w, xb.x, xb.y, xb.z, xb.w};

    // gelu(d) for d>=0:  d * sigmoid(2*alpha*(d + c*d^3)),  alpha=sqrt(2/pi)
    // exp via v_exp_f32:  e = exp2(kNeg * (d + c*d^3)),  g = d * rcp(1+e)
    const float kNeg = -(float)(2.0 * 0.7978845608028654 * 1.4426950408889634);
    const float kC   = 0.044715f;

#pragma unroll
    for (int j = 0; j < 4; ++j) {
        const int col   = n0 + j * 16 + lo;
        const float w2v = w2[col];
        float* outcol   = out + col;
#pragma unroll
        for (int r = 0; r < 8; ++r) {
            const int row = m0 + 8 * hi + r;
            float s    = x2r[r] + w2v;
            float d2   = fmaxf(fmaf(-2.0f, acc[j][r], s), 0.0f);
            float dist = __builtin_amdgcn_sqrtf(d2);
            float p    = fmaf(kC * dist, dist * dist, dist);   // d + c*d^3
            float e    = __builtin_amdgcn_exp2f(kNeg * p);     // in (0,1]
            float g    = dist * __builtin_amdgcn_rcpf(1.0f + e);
            __builtin_nontemporal_store(g, outcol + (size_t)row * WIDTH);
        }
    }
}

extern "C" void kernel_launch(void* const* d_in, const int* in_sizes, int n_in,
                              void* d_out, int out_size, void* d_ws, size_t ws_size,
                              hipStream_t stream) {
    const float* x = (const float*)d_in[0];   // [16384, 64]
    const float* w = (const float*)d_in[1];   // [4096, 64]
    float* out = (float*)d_out;               // [16384, 4096]

    float* x2 = (float*)d_ws;                 // 16384 floats
    float* w2 = x2 + BATCH;                   // 4096 floats (80 KB total)

    rownorm2_kernel<<<BATCH / 256, 256, 0, stream>>>(x, x2, BATCH);
    rownorm2_kernel<<<WIDTH / 256, 256, 0, stream>>>(w, w2, WIDTH);

    dim3 grid(WIDTH / 64, BATCH / 128);       // (64, 128) workgroups
    dist_gelu_wmma_kernel<<<grid, 256, 0, stream>>>(x, w, x2, w2, out);
}